// Attention_23467701305808
// MI455X (gfx1250) — compile-verified
//
#include <hip/hip_runtime.h>
#include <math.h>

typedef __attribute__((ext_vector_type(2))) float v2f;
typedef __attribute__((ext_vector_type(8))) float v8f;

#define N_MEL   80
#define N_KV    640
#define N_ATT   128
#define T_MEL   512
#define T_TEXT  256
#define BATCH   8

__device__ __forceinline__ float fast_tanh(float x) {
#if __has_builtin(__builtin_amdgcn_tanhf)
  return __builtin_amdgcn_tanhf(x);           // native gfx1250 v_tanh_f32 (TRANS)
#else
  return tanhf(x);
#endif
}

// -----------------------------------------------------------------------------
// Projection GEMM: out[b][t][a] = sum_c in[t*BATCH+b][c] * W[a][c]
// One wave per 16(row)x16(a) tile, f32 WMMA 16x16x4 over the C dimension.
// A-frag layout (ISA 7.12.2, 32-bit A 16x4): lanes 0-15 -> M=0..15, K=0,1;
//                                            lanes 16-31 -> M=0..15, K=2,3.
// B-frag mirrored: lane%16 = N, (lane/16)*2 = K base.
// -----------------------------------------------------------------------------
__global__ __launch_bounds__(32)
void proj_wmma(const float* __restrict__ in, const float* __restrict__ W,
               float* __restrict__ out, int T, int C) {
  const int lane = threadIdx.x;
  const int r0   = blockIdx.x * 16;          // tile over flattened (t*BATCH+b)
  const int a0   = blockIdx.y * 16;          // tile over A
  const int half = lane >> 4;                // 0 or 1
  const int kb   = half * 2;                 // K base within 4-step
  const int rl   = lane & 15;

  const float* arow = in + (size_t)(r0 + rl) * C + kb;   // A: row r0+rl
  const float* brow = W  + (size_t)(a0 + rl) * C + kb;   // B: col a0+rl (W row)

  v8f acc = {};
  for (int c = 0; c < C; c += 4) {
    v2f av = *(const v2f*)(arow + c);        // K=kb, kb+1 (8B aligned: C even)
    v2f bv = *(const v2f*)(brow + c);
    acc = __builtin_amdgcn_wmma_f32_16x16x4_f32(false, av, false, bv,
                                                (short)0, acc, false, false);
  }
  // D layout: lane col N=rl, VGPR i -> row M = i + half*8. row r -> (t,b)
#pragma unroll
  for (int i = 0; i < 8; ++i) {
    int r = r0 + i + half * 8;
    int t = r >> 3;                          // r / BATCH
    int b = r & 7;                           // r % BATCH
    out[((size_t)b * T + t) * N_ATT + a0 + rl] = acc[i];
  }
}

// -----------------------------------------------------------------------------
// Scores + softmax.  Block = (b, 16-row m-tile), 256 threads; thread tid owns
// text position t = tid.  sc[m] = sum_a v[a]*tanh(Q[m,a] + K[t,a]).
// Q tile in LDS (stride 132 floats: keeps float4 alignment, spreads banks);
// K row streamed through registers 16 floats at a time; 2048 v_tanh per thread.
// Block softmax: per-wave shfl_xor reduce -> LDS partials -> broadcast.
// -----------------------------------------------------------------------------
__global__ __launch_bounds__(256)
void score_softmax(const float* __restrict__ Q, const float* __restrict__ K,
                   const float* __restrict__ vvec,
                   float* __restrict__ attn, float* __restrict__ logp) {
  __shared__ float Qs[16 * 132];
  __shared__ float Vv[128];
  __shared__ float pmax[16][8];
  __shared__ float psum[16][8];

  const int tid  = threadIdx.x;
  const int b    = blockIdx.y;
  const int m0   = blockIdx.x * 16;
  const int lane = tid & 31;
  const int wave = tid >> 5;

  // stage Q tile: 16x128 floats = 512 float4, 2 per thread
#pragma unroll
  for (int i = 0; i < 2; ++i) {
    int e = 4 * (tid + 256 * i);
    int row = e >> 7, col = e & 127;
    *(float4*)&Qs[row * 132 + col] =
        *(const float4*)&Q[((size_t)b * T_MEL + m0 + row) * N_ATT + col];
  }
  if (tid < 32) *(float4*)&Vv[tid * 4] = *(const float4*)&vvec[tid * 4];
  __syncthreads();

  const float* Kp = K + ((size_t)b * T_TEXT + tid) * N_ATT;
  __builtin_prefetch(Kp, 0, 0);
  __builtin_prefetch(Kp + 64, 0, 0);

  float sc[16];
#pragma unroll
  for (int m = 0; m < 16; ++m) sc[m] = 0.f;

  for (int c = 0; c < N_ATT; c += 16) {
    float4 k0 = *(const float4*)(Kp + c);
    float4 k1 = *(const float4*)(Kp + c + 4);
    float4 k2 = *(const float4*)(Kp + c + 8);
    float4 k3 = *(const float4*)(Kp + c + 12);
    float4 w0 = *(const float4*)&Vv[c];
    float4 w1 = *(const float4*)&Vv[c + 4];
    float4 w2 = *(const float4*)&Vv[c + 8];
    float4 w3 = *(const float4*)&Vv[c + 12];
#pragma unroll
    for (int m = 0; m < 16; ++m) {
      const float4* q = (const float4*)&Qs[m * 132 + c];
      float4 q0 = q[0], q1 = q[1], q2 = q[2], q3 = q[3];
      float s = sc[m];
      s += w0.x * fast_tanh(q0.x + k0.x);  s += w0.y * fast_tanh(q0.y + k0.y);
      s += w0.z * fast_tanh(q0.z + k0.z);  s += w0.w * fast_tanh(q0.w + k0.w);
      s += w1.x * fast_tanh(q1.x + k1.x);  s += w1.y * fast_tanh(q1.y + k1.y);
      s += w1.z * fast_tanh(q1.z + k1.z);  s += w1.w * fast_tanh(q1.w + k1.w);
      s += w2.x * fast_tanh(q2.x + k2.x);  s += w2.y * fast_tanh(q2.y + k2.y);
      s += w2.z * fast_tanh(q2.z + k2.z);  s += w2.w * fast_tanh(q2.w + k2.w);
      s += w3.x * fast_tanh(q3.x + k3.x);  s += w3.y * fast_tanh(q3.y + k3.y);
      s += w3.z * fast_tanh(q3.z + k3.z);  s += w3.w * fast_tanh(q3.w + k3.w);
      sc[m] = s;
    }
  }

  // --- block-wide max over t per row m ---
#pragma unroll
  for (int m = 0; m < 16; ++m) {
    float x = sc[m];
    for (int off = 16; off > 0; off >>= 1) x = fmaxf(x, __shfl_xor(x, off, 32));
    if (lane == 0) pmax[m][wave] = x;
  }
  __syncthreads();
#pragma unroll
  for (int m = 0; m < 16; ++m) {
    float x = pmax[m][0];
#pragma unroll
    for (int w = 1; w < 8; ++w) x = fmaxf(x, pmax[m][w]);
    sc[m] = __expf(sc[m] - x);               // exp(score - rowmax)
  }
  // --- block-wide sum ---
#pragma unroll
  for (int m = 0; m < 16; ++m) {
    float x = sc[m];
    for (int off = 16; off > 0; off >>= 1) x += __shfl_xor(x, off, 32);
    if (lane == 0) psum[m][wave] = x;
  }
  __syncthreads();
#pragma unroll
  for (int m = 0; m < 16; ++m) {
    float s = psum[m][0];
#pragma unroll
    for (int w = 1; w < 8; ++w) s += psum[m][w];
    float a  = sc[m] * (1.0f / s);
    size_t idx = ((size_t)(b * T_MEL + m0 + m)) * T_TEXT + tid;  // coalesced
    attn[idx] = a;
    logp[idx] = __logf(a + 1e-8f);
  }
}

// -----------------------------------------------------------------------------
// Context GEMM: out[b][a][m] = sum_t attn[b][m][t] * V[b][t][a]
// One wave per 16m x 16a tile; K-loop over t in steps of 4 (f32 WMMA).
// Transposed store lands as two contiguous float4 per lane.
// -----------------------------------------------------------------------------
__global__ __launch_bounds__(32)
void ctx_wmma(const float* __restrict__ attn, const float* __restrict__ V,
              float* __restrict__ out) {
  const int lane = threadIdx.x;
  const int m0 = blockIdx.x * 16, a0 = blockIdx.y * 16, b = blockIdx.z;
  const int half = lane >> 4, kb = half * 2, rl = lane & 15;

  const float* arow = attn + ((size_t)b * T_MEL + m0 + rl) * T_TEXT + kb;
  const float* bcol = V + ((size_t)b * T_TEXT + kb) * N_ATT + a0 + rl;

  v8f acc = {};
  for (int t = 0; t < T_TEXT; t += 4) {
    v2f av = *(const v2f*)(arow + t);                 // attn[m][t+kb .. +1]
    v2f bv;
    bv.x = bcol[(size_t)t * N_ATT];                   // V[t+kb  ][a]
    bv.y = bcol[(size_t)(t + 1) * N_ATT];             // V[t+kb+1][a]
    acc = __builtin_amdgcn_wmma_f32_16x16x4_f32(false, av, false, bv,
                                                (short)0, acc, false, false);
  }
  // out[b][a0+rl][m0 + half*8 + i], i=0..7 contiguous -> two float4 stores
  float* op = out + ((size_t)b * N_ATT + a0 + rl) * T_MEL + m0 + half * 8;
  *(float4*)(op)     = make_float4(acc[0], acc[1], acc[2], acc[3]);
  *(float4*)(op + 4) = make_float4(acc[4], acc[5], acc[6], acc[7]);
}

// -----------------------------------------------------------------------------
extern "C" void kernel_launch(void* const* d_in, const int* in_sizes, int n_in,
                              void* d_out, int out_size, void* d_ws, size_t ws_size,
                              hipStream_t stream) {
  const float* queries = (const float*)d_in[0];   // [512,8,80]
  const float* keys    = (const float*)d_in[1];   // [256,8,640]
  const float* values  = (const float*)d_in[2];   // [256,8,640]
  const float* Wq      = (const float*)d_in[3];   // [128,80]
  const float* Wk      = (const float*)d_in[4];   // [128,640]
  const float* Wv      = (const float*)d_in[5];   // [128,640]
  const float* vvec    = (const float*)d_in[6];   // [128]

  float* ws = (float*)d_ws;
  float* Qp = ws;                       // [8][512][128] = 524288 f
  float* Kp = ws + 524288;              // [8][256][128] = 262144 f
  float* Vp = ws + 786432;              // [8][256][128] = 262144 f  (4 MB total)

  float* out  = (float*)d_out;          // [8][128][512]
  float* attn = out + 524288;           // [8][512][256]
  float* logp = attn + 1048576;         // [8][512][256]

  // projections (WMMA f32): rows = T*BATCH, cols = 128
  proj_wmma<<<dim3(256, 8), 32, 0, stream>>>(queries, Wq, Qp, T_MEL, N_MEL);
  proj_wmma<<<dim3(128, 8), 32, 0, stream>>>(keys,    Wk, Kp, T_TEXT, N_KV);
  proj_wmma<<<dim3(128, 8), 32, 0, stream>>>(values,  Wv, Vp, T_TEXT, N_KV);

  // additive scores + softmax (+ logprob) -- the compute-bound part
  score_softmax<<<dim3(T_MEL / 16, BATCH), 256, 0, stream>>>(Qp, Kp, vvec, attn, logp);

  // context GEMM (WMMA f32), transposed store to [B,A,M]
  ctx_wmma<<<dim3(T_MEL / 16, N_ATT / 16, BATCH), 32, 0, stream>>>(attn, Vp, out);
}